// atten_classifier_GAT_o_35734127902886
// MI455X (gfx1250) — compile-verified
//
#include <hip/hip_runtime.h>
#include <hip/hip_bf16.h>

// Shapes are fixed by the reference.
#define BB 32
#define NN 8192
#define DD 256
#define CC 1000

typedef __attribute__((ext_vector_type(2))) float v2f;
typedef __attribute__((ext_vector_type(4))) float v4f;
typedef __attribute__((ext_vector_type(8))) float v8f;

// ---------------------------------------------------------------------------
// k_zero: clear P/Q accumulators (they are atomically accumulated each call)
// ---------------------------------------------------------------------------
__global__ void k_zero(float* __restrict__ p, int n) {
    int i = blockIdx.x * blockDim.x + threadIdx.x;
    if (i < n) p[i] = 0.0f;
}

// ---------------------------------------------------------------------------
// k_v: v[d] = sum_k A1[k] * W[k,d]   (v = W^T A1), 1 block x 256 threads
// ---------------------------------------------------------------------------
__global__ __launch_bounds__(DD) void k_v(const float* __restrict__ A,
                                          const float* __restrict__ W,
                                          float* __restrict__ v) {
    const int d = threadIdx.x;
    float s = 0.0f;
    for (int k = 0; k < DD; ++k) s = fmaf(A[k], W[k * DD + d], s);
    v[d] = s;
}

// ---------------------------------------------------------------------------
// k_o1: o1 = obj @ W^T  (M=32, N=256, K=256) via V_WMMA_F32_16X16X4_F32.
// One wave per 16x16 output tile.  A[m,k]=obj[m,k], B[k,n]=W[n,k].
// f32 WMMA layouts (ISA 7.12.2): lane<16 holds K={0,1}, lane>=16 K={2,3};
// C/D vgpr j: M = j + 8*(lane/16), N = lane%16.
// ---------------------------------------------------------------------------
__global__ __launch_bounds__(32) void k_o1(const float* __restrict__ obj,
                                           const float* __restrict__ W,
                                           float* __restrict__ o1) {
    const int lane  = threadIdx.x;
    const int half  = lane >> 4;
    const int l16   = lane & 15;
    const int mbase = blockIdx.y * 16;
    const int nbase = blockIdx.x * 16;
    const int m     = mbase + l16;
    const int n     = nbase + l16;
    const int koff  = half * 2;

    v8f acc = {0.f, 0.f, 0.f, 0.f, 0.f, 0.f, 0.f, 0.f};
    for (int k = 0; k < DD; k += 4) {
        v2f a, b;
        a.x = obj[m * DD + k + koff];
        a.y = obj[m * DD + k + koff + 1];
        b.x = W[n * DD + k + koff];
        b.y = W[n * DD + k + koff + 1];
        acc = __builtin_amdgcn_wmma_f32_16x16x4_f32(false, a, false, b,
                                                    (short)0, acc, false, false);
    }
#pragma unroll
    for (int j = 0; j < 8; ++j) {
        const int mm = mbase + half * 8 + j;
        o1[mm * DD + nbase + l16] = acc[j];
    }
}

// ---------------------------------------------------------------------------
// k_c: c[b] = sum_k o1[b,k] * A2[k]     (A2 = A + 256)
// ---------------------------------------------------------------------------
__global__ __launch_bounds__(DD) void k_c(const float* __restrict__ o1,
                                          const float* __restrict__ A,
                                          float* __restrict__ c) {
    __shared__ float red[DD];
    const int b = blockIdx.x, t = threadIdx.x;
    red[t] = o1[b * DD + t] * A[DD + t];
    __syncthreads();
    for (int s = DD / 2; s > 0; s >>= 1) {
        if (t < s) red[t] += red[t + s];
        __syncthreads();
    }
    if (t == 0) c[b] = red[0];
}

// ---------------------------------------------------------------------------
// k_stream helper: consume one row (8 floats/lane), wave32 all-reduce the dot
// with v, accumulate P/Q partials in registers.
// ---------------------------------------------------------------------------
__device__ __forceinline__ void stream_row(const v4f x0, const v4f x1,
                                           const float vv[8],
                                           float accP[8], float accQ[8]) {
    const float xv[8] = {x0.x, x0.y, x0.z, x0.w, x1.x, x1.y, x1.z, x1.w};
    float t = 0.0f;
#pragma unroll
    for (int j = 0; j < 8; ++j) t = fmaf(xv[j], vv[j], t);
#pragma unroll
    for (int off = 16; off > 0; off >>= 1) t += __shfl_xor(t, off, 32);
#pragma unroll
    for (int j = 0; j < 8; ++j) {
        accP[j] = fmaf(t, xv[j], accP[j]);
        accQ[j] += xv[j];
    }
}

// ---------------------------------------------------------------------------
// k_stream: the bandwidth-bound pass over atom (268 MB, read once).
//   t      = v . atom[b,n,:]
//   P[b,d] += t * atom[b,n,d]
//   Q[b,d] += atom[b,n,d]
// grid = (32 chunks, 32 batches), block = 256 (8 waves). Each wave streams 32
// contiguous rows (1 KB each, 2x global_load_b128 NT loads/lane — atom is
// bigger than the 192 MB L2 and single-use, so TH_LOAD_NT avoids thrash).
// Software-pipelined: next row's loads issue before current row is consumed.
// wave32 shfl_xor all-reduce, register accumulation, LDS cross-wave combine,
// one f32 global atomic per column per block.
// ---------------------------------------------------------------------------
__global__ __launch_bounds__(256) void k_stream(const float* __restrict__ atom,
                                                const float* __restrict__ v,
                                                float* __restrict__ P,
                                                float* __restrict__ Q) {
    __shared__ float sP[8 * DD];
    __shared__ float sQ[8 * DD];

    const int b    = blockIdx.y;
    const int tid  = threadIdx.x;
    const int lane = tid & 31;
    const int wave = tid >> 5;
    const int col0 = lane * 8;          // 8 columns per lane

    // per-lane slice of v (kept in registers)
    const v4f* vp = reinterpret_cast<const v4f*>(v + col0);
    const v4f va = vp[0], vb = vp[1];
    const float vv[8] = {va.x, va.y, va.z, va.w, vb.x, vb.y, vb.z, vb.w};

    const float* base = atom + (size_t)b * NN * DD;
    const int rowBase = blockIdx.x * 256 + wave * 32;

    float accP[8] = {0.f, 0.f, 0.f, 0.f, 0.f, 0.f, 0.f, 0.f};
    float accQ[8] = {0.f, 0.f, 0.f, 0.f, 0.f, 0.f, 0.f, 0.f};

    // prologue: load row 0 (non-temporal b128 x2 per lane)
    const v4f* rp = reinterpret_cast<const v4f*>(
        base + (size_t)rowBase * DD + col0);
    v4f x0 = __builtin_nontemporal_load(rp);
    v4f x1 = __builtin_nontemporal_load(rp + 1);

    for (int i = 0; i < 31; ++i) {
        // issue next row's loads before consuming the current row
        const v4f* np = reinterpret_cast<const v4f*>(
            base + (size_t)(rowBase + i + 1) * DD + col0);
        const v4f y0 = __builtin_nontemporal_load(np);
        const v4f y1 = __builtin_nontemporal_load(np + 1);

        stream_row(x0, x1, vv, accP, accQ);
        x0 = y0;
        x1 = y1;
    }
    stream_row(x0, x1, vv, accP, accQ);   // epilogue: last row

    // cross-wave combine in LDS, then one atomic per column per block
#pragma unroll
    for (int j = 0; j < 8; ++j) {
        sP[wave * DD + col0 + j] = accP[j];
        sQ[wave * DD + col0 + j] = accQ[j];
    }
    __syncthreads();

    float p = 0.0f, q = 0.0f;
#pragma unroll
    for (int w = 0; w < 8; ++w) {
        p += sP[w * DD + tid];
        q += sQ[w * DD + tid];
    }
    atomicAdd(&P[b * DD + tid], p);
    atomicAdd(&Q[b * DD + tid], q);
}

// ---------------------------------------------------------------------------
// k_msg: msg[b,n] = sum_d (P[b,d] + c[b]*Q[b,d]) * W[n,d]   (WMMA f32)
// The s = P + c*Q fusion happens in the A-fragment load.
// ---------------------------------------------------------------------------
__global__ __launch_bounds__(32) void k_msg(const float* __restrict__ P,
                                            const float* __restrict__ Q,
                                            const float* __restrict__ c,
                                            const float* __restrict__ W,
                                            float* __restrict__ msg) {
    const int lane  = threadIdx.x;
    const int half  = lane >> 4;
    const int l16   = lane & 15;
    const int mbase = blockIdx.y * 16;
    const int nbase = blockIdx.x * 16;
    const int m     = mbase + l16;
    const int n     = nbase + l16;
    const int koff  = half * 2;
    const float cb  = c[m];

    v8f acc = {0.f, 0.f, 0.f, 0.f, 0.f, 0.f, 0.f, 0.f};
    for (int k = 0; k < DD; k += 4) {
        const int j = k + koff;
        v2f a, b;
        a.x = fmaf(cb, Q[m * DD + j],     P[m * DD + j]);
        a.y = fmaf(cb, Q[m * DD + j + 1], P[m * DD + j + 1]);
        b.x = W[n * DD + j];
        b.y = W[n * DD + j + 1];
        acc = __builtin_amdgcn_wmma_f32_16x16x4_f32(false, a, false, b,
                                                    (short)0, acc, false, false);
    }
#pragma unroll
    for (int j = 0; j < 8; ++j) {
        const int mm = mbase + half * 8 + j;
        msg[mm * DD + nbase + l16] = acc[j];
    }
}

// ---------------------------------------------------------------------------
// k_preds: preds = [o1|msg] @ CW^T + CB  (M=32, N=1000 (tiled to 1008), K=512)
// WMMA f32; out-of-range class columns are clamped for loads (EXEC must stay
// all-ones for WMMA) and skipped on store.
// ---------------------------------------------------------------------------
__global__ __launch_bounds__(32) void k_preds(const float* __restrict__ o1,
                                              const float* __restrict__ msg,
                                              const float* __restrict__ CW,
                                              const float* __restrict__ CB,
                                              float* __restrict__ out) {
    const int lane  = threadIdx.x;
    const int half  = lane >> 4;
    const int l16   = lane & 15;
    const int mbase = blockIdx.y * 16;
    const int nbase = blockIdx.x * 16;
    const int m     = mbase + l16;
    const int n     = nbase + l16;
    const int nc    = (n < CC) ? n : (CC - 1);
    const int koff  = half * 2;

    v8f acc = {0.f, 0.f, 0.f, 0.f, 0.f, 0.f, 0.f, 0.f};
    for (int k = 0; k < 2 * DD; k += 4) {
        const int j = k + koff;          // j and j+1 never straddle DD (k%4==0)
        const float* xr = (j < DD) ? (o1 + m * DD) : (msg + m * DD - DD);
        v2f a, b;
        a.x = xr[j];
        a.y = xr[j + 1];
        b.x = CW[nc * (2 * DD) + j];
        b.y = CW[nc * (2 * DD) + j + 1];
        acc = __builtin_amdgcn_wmma_f32_16x16x4_f32(false, a, false, b,
                                                    (short)0, acc, false, false);
    }
#pragma unroll
    for (int j = 0; j < 8; ++j) {
        const int mm = mbase + half * 8 + j;
        const int nn = nbase + l16;
        if (nn < CC) out[mm * CC + nn] = acc[j] + CB[nn];
    }
}

// ---------------------------------------------------------------------------
// Launch
// ---------------------------------------------------------------------------
extern "C" void kernel_launch(void* const* d_in, const int* in_sizes, int n_in,
                              void* d_out, int out_size, void* d_ws, size_t ws_size,
                              hipStream_t stream) {
    const float* atom = (const float*)d_in[0];   // (B,N,D)
    const float* obj  = (const float*)d_in[1];   // (B,D)
    const float* W    = (const float*)d_in[2];   // (D,D)
    const float* A    = (const float*)d_in[3];   // (1,2D)
    const float* CW   = (const float*)d_in[4];   // (C,2D)
    const float* CB   = (const float*)d_in[5];   // (C,)
    float* out = (float*)d_out;                  // (B,C)

    float* ws  = (float*)d_ws;
    float* v   = ws;                  // 256
    float* o1  = ws + 256;            // 32*256
    float* msg = ws + 8448;           // 32*256
    float* c   = ws + 16640;          // 32
    float* P   = ws + 16672;          // 32*256
    float* Q   = ws + 24864;          // 32*256  (contiguous after P)

    k_zero<<<dim3(64), 256, 0, stream>>>(P, 2 * BB * DD);
    k_v<<<dim3(1), DD, 0, stream>>>(A, W, v);
    k_o1<<<dim3(DD / 16, BB / 16), 32, 0, stream>>>(obj, W, o1);
    k_c<<<dim3(BB), DD, 0, stream>>>(o1, A, c);
    k_stream<<<dim3(NN / 256, BB), 256, 0, stream>>>(atom, v, P, Q);
    k_msg<<<dim3(DD / 16, BB / 16), 32, 0, stream>>>(P, Q, c, W, msg);
    k_preds<<<dim3((CC + 15) / 16, BB / 16), 32, 0, stream>>>(o1, msg, CW, CB, out);
}